// EncodeProcessDecode_84293028151463
// MI455X (gfx1250) — compile-verified
//
#include <hip/hip_runtime.h>
#include <hip/hip_bf16.h>
#include <cstddef>

#define HID 128

typedef float v2f __attribute__((ext_vector_type(2)));
typedef float v8f __attribute__((ext_vector_type(8)));

// ---- order-preserving float <-> u32 mapping for atomic min ----
__device__ __forceinline__ unsigned fmap(float f) {
    unsigned b = __float_as_uint(f);
    return (b & 0x80000000u) ? ~b : (b | 0x80000000u);
}
__device__ __forceinline__ float funmap(unsigned m) {
    unsigned b = (m & 0x80000000u) ? (m & 0x7FFFFFFFu) : ~m;
    return __uint_as_float(b);
}
#define MAPPED_POS_INF 0xFF800000u   // fmap(+inf)

// =====================================================================
// Kernel 1: h = relu(x @ W_enc + b_enc)
// block = 128 thr (4 waves); wave w -> 16x16 tile at cols colBase + 16w
// =====================================================================
__global__ __launch_bounds__(128) void enc_kernel(
    const float* __restrict__ x, const float* __restrict__ W,
    const float* __restrict__ b, float* __restrict__ h, int n)
{
    __shared__ __align__(16) float Ws[HID * 64];   // W[:, colBase..colBase+63]  32 KB
    const int colBase = blockIdx.y * 64;
    const int rowBase = blockIdx.x * 16;

    for (int i = threadIdx.x; i < HID * 64; i += 128) {
        int k = i >> 6, c = i & 63;
        Ws[i] = W[k * HID + colBase + c];
    }
    __syncthreads();

    const int wave = threadIdx.x >> 5;
    const int lane = threadIdx.x & 31;
    const int l15  = lane & 15;
    const int koff = (lane >> 4) << 1;          // 0 or 2 (ISA fp32 A layout)
    const int col  = wave * 16 + l15;           // within 64-col slice

    int arow = rowBase + l15; if (arow >= n) arow = n - 1;
    const float* xrow = x + (size_t)arow * HID;

    v8f acc = {};
    for (int k = 0; k < HID; k += 4) {
        v2f a, bm;
        a.x  = xrow[k + koff];
        a.y  = xrow[k + koff + 1];
        bm.x = Ws[(k + koff) * 64 + col];
        bm.y = Ws[(k + koff + 1) * 64 + col];
        acc = __builtin_amdgcn_wmma_f32_16x16x4_f32(false, a, false, bm,
                                                    (short)0, acc, false, false);
    }
    const float bias  = b[colBase + col];
    const int   rbase = rowBase + ((lane >> 4) << 3);
    float* hout = h + (size_t)rbase * HID + colBase + col;
    if (rowBase + 16 <= n) {                    // uniform fast path: no per-row EXEC churn
        #pragma unroll
        for (int r = 0; r < 8; ++r) {
            float v = acc[r] + bias;
            hout[(size_t)r * HID] = v > 0.f ? v : 0.f;
        }
    } else {
        #pragma unroll
        for (int r = 0; r < 8; ++r) {
            if (rbase + r < n) {
                float v = acc[r] + bias;
                hout[(size_t)r * HID] = v > 0.f ? v : 0.f;
            }
        }
    }
}

// =====================================================================
// Kernel 2: agg keys <- mapped(+inf)
// =====================================================================
__global__ __launch_bounds__(256) void init_kernel(unsigned* __restrict__ agg, int total)
{
    int i = blockIdx.x * 256 + threadIdx.x;
    if (i < total) agg[i] = MAPPED_POS_INF;
}

// =====================================================================
// Kernel 3: fused message GEMM + scatter-min
// block = 128 thr (4 waves); wave -> 16 edges x 128 features.
// K outermost with 8 live accumulator tiles: A loaded once per k-step,
// 8 independent WMMA chains keep the matrix pipe full.
// =====================================================================
__global__ __launch_bounds__(128) void msg_kernel(
    const float* __restrict__ h, const int* __restrict__ edge_index,
    const float* __restrict__ edge_attr, const float* __restrict__ Wm,
    const float* __restrict__ bmv, unsigned* __restrict__ agg, int E)
{
    __shared__ __align__(16) float hbuf[4][16][HID];   // 32 KB, gathered src rows
    const int wave  = threadIdx.x >> 5;
    const int lane  = threadIdx.x & 31;
    const int eBase = (blockIdx.x * 4 + wave) * 16;

    int s = 0, d = 0; float ea = 0.f;
    if (lane < 16) {
        int e = eBase + lane;
        if (e < E) {
            s  = edge_index[e];
            d  = edge_index[(size_t)E + e];
            ea = edge_attr[e];
        }
    }

    // gather 16 rows of h (float4-wide); same-wave DS ops are in-order
    #pragma unroll 4
    for (int i = 0; i < 16; ++i) {
        int idx  = lane + i * 32;            // 512 float4 slots
        int row  = idx >> 5;
        int c4   = idx & 31;
        int srow = __shfl(s, row);
        float4 v = ((const float4*)(h + (size_t)srow * HID))[c4];
        ((float4*)&hbuf[wave][row][0])[c4] = v;
    }

    const int l15  = lane & 15;
    const int koff = (lane >> 4) << 1;
    const float* hA = &hbuf[wave][l15][0];

    v8f acc[8] = {};
    for (int k = 0; k < HID; k += 4) {
        v2f a;
        a.x = hA[k + koff];
        a.y = hA[k + koff + 1];
        const float* w0 = Wm + (size_t)(k + koff)     * HID + l15;
        const float* w1 = Wm + (size_t)(k + koff + 1) * HID + l15;
        #pragma unroll
        for (int ct = 0; ct < 8; ++ct) {
            v2f bm;
            bm.x = w0[ct * 16];
            bm.y = w1[ct * 16];
            acc[ct] = __builtin_amdgcn_wmma_f32_16x16x4_f32(false, a, false, bm,
                                                            (short)0, acc[ct], false, false);
        }
    }

    // broadcast per-edge scalars once
    float eav[8]; int dn[8];
    #pragma unroll
    for (int r = 0; r < 8; ++r) {
        int eslot = r + ((lane >> 4) << 3);
        eav[r] = __shfl(ea, eslot);
        dn[r]  = __shfl(d,  eslot);
    }

    const bool full = (eBase + 16 <= E);       // wave-uniform
    #pragma unroll
    for (int ct = 0; ct < 8; ++ct) {
        const int   col   = ct * 16 + l15;
        const float wlast = Wm[(size_t)HID * HID + col];  // edge_attr row of W_msg
        const float bias  = bmv[col];
        if (full) {
            #pragma unroll
            for (int r = 0; r < 8; ++r) {
                float v = acc[ct][r] + eav[r] * wlast + bias;
                atomicMin(&agg[(size_t)dn[r] * HID + col], fmap(v));  // global_atomic_min_u32
            }
        } else {
            #pragma unroll
            for (int r = 0; r < 8; ++r) {
                int eslot = r + ((lane >> 4) << 3);
                if (eBase + eslot < E) {
                    float v = acc[ct][r] + eav[r] * wlast + bias;
                    atomicMin(&agg[(size_t)dn[r] * HID + col], fmap(v));
                }
            }
        }
    }
}

// =====================================================================
// Kernel 4: u = concat(h, agg) @ W_upd + b_upd ; out = sigmoid(u @ W_dec + b_dec)
// block = 256 thr (8 waves); block -> 16 rows x 128 cols, K = 256
// =====================================================================
__global__ __launch_bounds__(256) void upd_dec_kernel(
    const float* __restrict__ h, const unsigned* __restrict__ agg,
    const float* __restrict__ Wu, const float* __restrict__ bu,
    const float* __restrict__ Wd, const float* __restrict__ bd,
    float* __restrict__ out, int n)
{
    __shared__ __align__(16) float Atile[16][256];   // 16 KB concat(h, agg)
    __shared__ __align__(16) float utile[16][HID];   //  8 KB updated features
    const int rowBase = blockIdx.x * 16;

    for (int i = threadIdx.x; i < 16 * 256; i += 256) {
        int r = i >> 8, c = i & 255;
        int row = rowBase + r; if (row >= n) row = n - 1;
        float v;
        if (c < HID) {
            v = h[(size_t)row * HID + c];
        } else {
            unsigned m = agg[(size_t)row * HID + (c - HID)];
            v = (m == MAPPED_POS_INF) ? 0.f : funmap(m);   // isinf -> 0
        }
        Atile[r][c] = v;
    }
    __syncthreads();

    const int wave = threadIdx.x >> 5;
    const int lane = threadIdx.x & 31;
    const int l15  = lane & 15;
    const int koff = (lane >> 4) << 1;
    const int col  = wave * 16 + l15;

    v8f acc = {};
    for (int k = 0; k < 256; k += 4) {
        v2f a, bm;
        a.x  = Atile[l15][k + koff];
        a.y  = Atile[l15][k + koff + 1];
        bm.x = Wu[(size_t)(k + koff) * HID + col];
        bm.y = Wu[(size_t)(k + koff + 1) * HID + col];
        acc = __builtin_amdgcn_wmma_f32_16x16x4_f32(false, a, false, bm,
                                                    (short)0, acc, false, false);
    }
    const float bias = bu[col];
    #pragma unroll
    for (int r = 0; r < 8; ++r)
        utile[r + ((lane >> 4) << 3)][col] = acc[r] + bias;
    __syncthreads();

    // decoder: 16 rows x 8 partial dots (threads 0..127), wave-local reduction
    if (threadIdx.x < 128) {
        int row = threadIdx.x >> 3, part = threadIdx.x & 7;
        float sum = 0.f;
        #pragma unroll
        for (int j = 0; j < 16; ++j)
            sum += utile[row][part * 16 + j] * Wd[part * 16 + j];
        sum += __shfl_down(sum, 4);
        sum += __shfl_down(sum, 2);
        sum += __shfl_down(sum, 1);
        if (part == 0) {
            int orow = rowBase + row;
            if (orow < n) out[orow] = 1.f / (1.f + __expf(-(sum + bd[0])));
        }
    }
}

// =====================================================================
extern "C" void kernel_launch(void* const* d_in, const int* in_sizes, int n_in,
                              void* d_out, int out_size, void* d_ws, size_t ws_size,
                              hipStream_t stream)
{
    const float* x     = (const float*)d_in[0];
    const int*   ei    = (const int*)  d_in[1];   // edge_index (int32 under JAX defaults)
    const float* ea    = (const float*)d_in[2];
    const float* W_enc = (const float*)d_in[3];
    const float* b_enc = (const float*)d_in[4];
    const float* W_msg = (const float*)d_in[5];
    const float* b_msg = (const float*)d_in[6];
    const float* W_upd = (const float*)d_in[7];
    const float* b_upd = (const float*)d_in[8];
    const float* W_dec = (const float*)d_in[9];
    const float* b_dec = (const float*)d_in[10];

    const int n = in_sizes[0] / HID;
    const int E = in_sizes[1] / 2;

    float*    h   = (float*)d_ws;
    unsigned* agg = (unsigned*)((char*)d_ws + (size_t)n * HID * sizeof(float));
    float*    o   = (float*)d_out;

    const int rowTiles = (n + 15) / 16;

    dim3 gEnc(rowTiles, 2);
    enc_kernel<<<gEnc, 128, 0, stream>>>(x, W_enc, b_enc, h, n);

    int total = n * HID;
    init_kernel<<<(total + 255) / 256, 256, 0, stream>>>(agg, total);

    msg_kernel<<<(E + 63) / 64, 128, 0, stream>>>(h, ei, ea, W_msg, b_msg, agg, E);

    upd_dec_kernel<<<rowTiles, 256, 0, stream>>>(h, agg, W_upd, b_upd, W_dec, b_dec, o, n);
}